// DeltaEncoder_44092134260942
// MI455X (gfx1250) — compile-verified
//
#include <hip/hip_runtime.h>
#include <stdint.h>

// x[32,512,1000] f32, sequential scan over last axis. Rows = 32*512 = 16384.
#define T_       1000
#define ROWS     32      // rows per block == threads per block (one wave)
#define NT       32
#define CT       32      // main-chunk timesteps: 32 f32 = 128 B/row
#define NMAIN    31      // 31*32 = 992
#define CTT      8       // tail timesteps: 992 + 8 = 1000
#define INPITCH  33      // TDM pad: +1 dword per 32 dwords -> odd pitch, 0 bank conflicts
#define OUTPITCH 36      // 144 B rows: 16B-aligned for b128 async stores, 2-way banks

typedef uint32_t u32x4 __attribute__((ext_vector_type(4)));
typedef uint32_t u32x8 __attribute__((ext_vector_type(8)));

// ---- CDNA5 Tensor Data Mover: one descriptor moves a whole 2D tile --------
// D# group0: {count=1, lds_addr, global_addr[31:0], global_addr[56:32]|type=2}
// D# group1: data_size=4B, pad_enable, pad_interval=32dw, pad_amount=1dw,
//            tensor_dim0=tileX, tensor_dim1=ROWS, tile_dim0=tileX,
//            tile_dim1=ROWS, tensor_dim0_stride=T_
__device__ __forceinline__ void tdm_load_tile(uint32_t lds_off, const void* g,
                                              uint32_t tileX) {
    const uint64_t ga = (uint64_t)(uintptr_t)g;
    u32x4 g0;
    g0[0] = 1u;                                               // count=1 (valid)
    g0[1] = lds_off;                                          // lds_addr (bytes)
    g0[2] = (uint32_t)ga;                                     // global_addr lo
    g0[3] = (uint32_t)((ga >> 32) & 0x1FFFFFFu) | 0x80000000u;// ga[56:32]|type=2
    u32x8 g1;
    g1[0] = (2u << 16) | (1u << 20) | (4u << 22);             // 4B, pad en, every 32dw, +1dw
    g1[1] = tileX << 16;                                      // tensor_dim0[15:0] @ bits63:48
    g1[2] = (uint32_t)ROWS << 16;                             // tensor_dim1[15:0] @ bits95:80
    g1[3] = tileX << 16;                                      // tile_dim0 @ bits127:112
    g1[4] = (uint32_t)ROWS;                                   // tile_dim1 @ bits143:128
    g1[5] = (uint32_t)T_;                                     // tensor_dim0_stride[31:0]
    g1[6] = 0u;
    g1[7] = 0u;
    asm volatile("tensor_load_to_lds %0, %1, null, null"
                 :: "s"(g0), "s"(g1) : "memory");
}

// ---- CDNA5 async LDS->global store (ASYNCcnt) -----------------------------
__device__ __forceinline__ void async_store_b128(void* g, uint32_t lds_off) {
    asm volatile("global_store_async_from_lds_b128 %0, %1, off"
                 :: "v"((uint64_t)(uintptr_t)g), "v"(lds_off) : "memory");
}
__device__ __forceinline__ void wait_async_le(int) = delete;
__device__ __forceinline__ void wait_async8() {
    asm volatile("s_wait_asynccnt 0x8" ::: "memory");  // allow newest batch in flight
}
__device__ __forceinline__ void wait_async0() {
    asm volatile("s_wait_asynccnt 0x0" ::: "memory");
}
// ---------------------------------------------------------------------------

__global__ __launch_bounds__(NT)
void DeltaEncoder_kernel(const float* __restrict__ x, float* __restrict__ out) {
    __shared__ __align__(16) float itile[2][ROWS * INPITCH];  // TDM-padded input
    __shared__ __align__(16) float otile[2][ROWS * OUTPITCH]; // spike staging

    const int       tid     = threadIdx.x;
    const long long rowBase = (long long)blockIdx.x * ROWS;
    const float*    xblk    = x   + rowBase * T_;
    float*          oblk    = out + rowBase * T_;

    float prev = 0.0f, acc = 0.0f;  // per-row recurrence state

    // Prefetch chunk 0
    tdm_load_tile((uint32_t)(uintptr_t)&itile[0][0], xblk, CT);

    for (int c = 0; c < NMAIN; ++c) {
        const int cur = c & 1, nxt = cur ^ 1;

        // Prefetch next tile (TDM runs during the scan below)
        if (c + 1 < NMAIN)
            tdm_load_tile((uint32_t)(uintptr_t)&itile[nxt][0],
                          xblk + (c + 1) * CT, CT);
        else
            tdm_load_tile((uint32_t)(uintptr_t)&itile[nxt][0],
                          xblk + NMAIN * CT, CTT);  // tail tile

        __builtin_amdgcn_s_wait_tensorcnt(1);  // chunk c resident; next may fly
        wait_async8();                         // otile[cur] drained (chunk c-2)

        // Scan CT steps: conflict-free reads (pitch 33), spikes to otile
        const float* irow = &itile[cur][tid * INPITCH];
        float*       orow = &otile[cur][tid * OUTPITCH];
        #pragma unroll
        for (int tt = 0; tt < CT; ++tt) {
            const float xv    = irow[tt];
            const float delta = xv - prev;
            acc = acc * 0.9f + delta;
            const bool p = acc >  0.1f;
            const bool n = acc < -0.1f;
            orow[tt] = p ? 1.0f : (n ? -1.0f : 0.0f);
            if (p || n) acc = 0.0f;
            prev = xv;
        }

        // Async drain chunk c: 8 x b128 per thread, coalesced 4-row runs
        float* gbase = oblk + c * CT;
        #pragma unroll
        for (int k = 0; k < 8; ++k) {
            const int j    = tid + k * NT;
            const int lrow = j >> 3;
            const int seg  = j & 7;
            async_store_b128(gbase + (long long)lrow * T_ + seg * 4,
                             (uint32_t)(uintptr_t)&otile[cur][lrow * OUTPITCH + seg * 4]);
        }
    }

    // ---- tail chunk (8 steps), buffer = NMAIN&1 = 1 ----
    {
        const int cur = NMAIN & 1;
        __builtin_amdgcn_s_wait_tensorcnt(0);  // tail tile resident
        wait_async8();                         // otile[cur] free (chunk 29 done)

        // TDM pad fires every 32 dwords = every 4 tail rows of 8 dwords
        const float* irow = &itile[cur][tid * CTT + (tid >> 2)];
        float*       orow = &otile[cur][tid * OUTPITCH];
        #pragma unroll
        for (int tt = 0; tt < CTT; ++tt) {
            const float xv    = irow[tt];
            const float delta = xv - prev;
            acc = acc * 0.9f + delta;
            const bool p = acc >  0.1f;
            const bool n = acc < -0.1f;
            orow[tt] = p ? 1.0f : (n ? -1.0f : 0.0f);
            if (p || n) acc = 0.0f;
            prev = xv;
        }

        float* gbase = oblk + NMAIN * CT;
        #pragma unroll
        for (int k = 0; k < 2; ++k) {
            const int j    = tid + k * NT;
            const int lrow = j >> 1;
            const int seg  = j & 1;
            async_store_b128(gbase + (long long)lrow * T_ + seg * 4,
                             (uint32_t)(uintptr_t)&otile[cur][lrow * OUTPITCH + seg * 4]);
        }
    }

    wait_async0();  // drain final stores
}

extern "C" void kernel_launch(void* const* d_in, const int* in_sizes, int n_in,
                              void* d_out, int out_size, void* d_ws, size_t ws_size,
                              hipStream_t stream) {
    (void)n_in; (void)d_ws; (void)ws_size; (void)in_sizes;
    const float* x   = (const float*)d_in[0];
    float*       out = (float*)d_out;

    const int totalRows = out_size / T_;     // 16384
    const int blocks    = totalRows / ROWS;  // 512

    DeltaEncoder_kernel<<<blocks, NT, 0, stream>>>(x, out);
}